// RobertaHybridEncoder_55301998903835
// MI455X (gfx1250) — compile-verified
//
#include <hip/hip_runtime.h>
#include <hip/hip_bf16.h>
#include <float.h>

#define B_    4
#define S_    512
#define H_    768
#define D_    24
#define M_    96
#define OUTH_ 768
#define K3H_  2304          // 3*H
#define ROWS_ (B_*S_)       // 2048

typedef float v2f __attribute__((ext_vector_type(2)));
typedef float v8f __attribute__((ext_vector_type(8)));

__device__ __forceinline__ v8f wmma_f32(v2f a, v2f b, v8f c) {
    // V_WMMA_F32_16X16X4_F32: D = A(16x4) * B(4x16) + C, fp32 all the way
    return __builtin_amdgcn_wmma_f32_16x16x4_f32(
        false, a, false, b, (short)0, c, false, false);
}

// LDS panel geometry for the tiled GEMMs (128x64 block tile, KC=32)
#define KC_    32
#define BM_    128
#define BN_    64
#define LDA_   (KC_ + 1)    // 33: pad to avoid stride-32 bank conflicts
#define LDB_   (BN_ + 1)    // 65: pad to avoid stride-64 bank conflicts

// ---------------------------------------------------------------------------
// Kernel 1: projections  Zj = H_j @ p_j_w^T,  Zi = H_i @ p_i_w^T   (K=768, tiny)
// ---------------------------------------------------------------------------
__global__ void k_proj(const float* __restrict__ Hj, const float* __restrict__ Hi,
                       const float* __restrict__ pjw, const float* __restrict__ piw,
                       float* __restrict__ Zj, float* __restrict__ Zi) {
    int idx = blockIdx.x * blockDim.x + threadIdx.x;
    const int total = ROWS_ * D_;
    if (idx >= 2 * total) return;
    int which = idx / total;
    int rem   = idx % total;
    int row   = rem / D_;
    int d     = rem % D_;
    const float* x = (which == 0 ? Hj : Hi) + (size_t)row * H_;
    const float* w = (which == 0 ? pjw : piw) + (size_t)d * H_;
    float acc = 0.f;
    #pragma unroll 8
    for (int h = 0; h < H_; ++h) acc += x[h] * w[h];
    (which == 0 ? Zj : Zi)[(size_t)row * D_ + d] = acc;
}

// ---------------------------------------------------------------------------
// Kernel 2: aj[row][m] = Zj[row]·Wa[m],  bi[row][m] = Zi[row]·Wb[m]
// ---------------------------------------------------------------------------
__global__ void k_ab(const float* __restrict__ Zj, const float* __restrict__ Zi,
                     const float* __restrict__ sw1,
                     float* __restrict__ aj, float* __restrict__ bi) {
    int idx = blockIdx.x * blockDim.x + threadIdx.x;
    const int total = ROWS_ * M_;
    if (idx >= 2 * total) return;
    int which = idx / total;
    int rem   = idx % total;
    int row   = rem / M_;
    int m     = rem % M_;
    const float* z = (which == 0 ? Zj : Zi) + (size_t)row * D_;
    const float* w = sw1 + (size_t)m * (4 * D_) + (which == 0 ? 0 : D_);
    float acc = 0.f;
    #pragma unroll
    for (int d = 0; d < D_; ++d) acc += z[d] * w[d];
    (which == 0 ? aj : bi)[(size_t)row * M_ + m] = acc;
}

// ---------------------------------------------------------------------------
// Kernel 3: fused pair-scorer + softmax (unchanged structure: WMMA-dense,
// B panel in LDS, A frag built in-registers and reused over 6 n-tiles).
// ---------------------------------------------------------------------------
__global__ void __launch_bounds__(256)
k_score(const float* __restrict__ Zj, const float* __restrict__ Zi,
        const float* __restrict__ aj, const float* __restrict__ bi_g,
        const float* __restrict__ sw1, const float* __restrict__ sb1,
        const float* __restrict__ sw2, const float* __restrict__ sb2,
        const float* __restrict__ mask, float* __restrict__ probs) {
    __shared__ float zjs[D_];
    __shared__ float ajs[M_];
    __shared__ float lb[48 * M_];     // lb[k][m] = s_w1[m][48+k]  (Wc|Wd)
    __shared__ float logits[S_];
    __shared__ float red[256];

    const int row = blockIdx.x;       // b*S + s
    const int b   = row / S_;
    const int tid = threadIdx.x;

    for (int i = tid; i < D_;      i += 256) zjs[i] = Zj[(size_t)row * D_ + i];
    for (int i = tid; i < M_;      i += 256) ajs[i] = aj[(size_t)row * M_ + i];
    for (int i = tid; i < 48 * M_; i += 256) {
        int k = i / M_, m = i % M_;
        lb[i] = sw1[(size_t)m * (4 * D_) + 48 + k];
    }
    __syncthreads();

    const int wave = tid >> 5;
    const int lane = tid & 31;
    const int half = lane >> 4;
    const int r    = lane & 15;

    float w2r[6], b1r[6];
    #pragma unroll
    for (int nt = 0; nt < 6; ++nt) { int m = nt * 16 + r; w2r[nt] = sw2[m]; b1r[nt] = sb1[m]; }

    for (int tt = wave; tt < S_ / 16; tt += 8) {
        const int t0 = tt * 16;
        const float* zir = Zi + (size_t)(b * S_ + t0 + r) * D_;

        v8f c[6];
        #pragma unroll
        for (int nt = 0; nt < 6; ++nt)
            #pragma unroll
            for (int i = 0; i < 8; ++i) c[nt][i] = 0.f;

        #pragma unroll
        for (int k = 0; k < 48; k += 4) {
            const int kk = k + 2 * half;
            v2f a;
            if (kk < D_) { a.x = zjs[kk] * zir[kk];           a.y = zjs[kk + 1] * zir[kk + 1]; }
            else { int d0 = kk - D_;
                   a.x = fabsf(zjs[d0] - zir[d0]);   a.y = fabsf(zjs[d0 + 1] - zir[d0 + 1]); }
            #pragma unroll
            for (int nt = 0; nt < 6; ++nt) {
                v2f bb;
                bb.x = lb[(size_t)kk * M_ + nt * 16 + r];
                bb.y = lb[(size_t)(kk + 1) * M_ + nt * 16 + r];
                c[nt] = wmma_f32(a, bb, c[nt]);
            }
        }

        float acc[8];
        #pragma unroll
        for (int i = 0; i < 8; ++i) acc[i] = 0.f;
        #pragma unroll
        for (int nt = 0; nt < 6; ++nt) {
            const int m = nt * 16 + r;
            #pragma unroll
            for (int i = 0; i < 8; ++i) {
                const int t = t0 + i + 8 * half;
                float v = c[nt][i] + ajs[m] + bi_g[(size_t)(b * S_ + t) * M_ + m] + b1r[nt];
                v = v > 0.f ? v : 0.f;
                acc[i] += v * w2r[nt];
            }
        }
        #pragma unroll
        for (int i = 0; i < 8; ++i) {
            acc[i] += __shfl_xor(acc[i], 1, 32);
            acc[i] += __shfl_xor(acc[i], 2, 32);
            acc[i] += __shfl_xor(acc[i], 4, 32);
            acc[i] += __shfl_xor(acc[i], 8, 32);
        }
        if (r == 0) {
            #pragma unroll
            for (int i = 0; i < 8; ++i)
                logits[t0 + i + 8 * half] = acc[i] + sb2[0];
        }
    }
    __syncthreads();

    float lmax = -FLT_MAX;
    for (int t = tid; t < S_; t += 256) {
        float v = logits[t] + (1.0f - mask[(size_t)b * S_ + t]) * (-3.4028234663852886e38f);
        logits[t] = v;
        lmax = fmaxf(lmax, v);
    }
    red[tid] = lmax; __syncthreads();
    for (int off = 128; off > 0; off >>= 1) {
        if (tid < off) red[tid] = fmaxf(red[tid], red[tid + off]);
        __syncthreads();
    }
    const float mx = red[0];
    __syncthreads();
    float lsum = 0.f;
    for (int t = tid; t < S_; t += 256) {
        float e = __expf(logits[t] - mx);
        logits[t] = e;
        lsum += e;
    }
    red[tid] = lsum; __syncthreads();
    for (int off = 128; off > 0; off >>= 1) {
        if (tid < off) red[tid] += red[tid + off];
        __syncthreads();
    }
    const float inv = 1.0f / red[0];
    for (int t = tid; t < S_; t += 256)
        probs[(size_t)row * S_ + t] = logits[t] * inv;
}

// ---------------------------------------------------------------------------
// Kernel 4: ctx[b] = probs[b](512x512) @ H_i[b](512x768)
// LDS-staged 128x64 block tile, 8 waves x (16x64) register tiles.
// ---------------------------------------------------------------------------
__global__ void __launch_bounds__(256)
k_ctx(const float* __restrict__ probs, const float* __restrict__ Hi,
      float* __restrict__ ctx) {
    __shared__ float As[BM_ * LDA_];
    __shared__ float Bs[KC_ * LDB_];

    const int tilesN = H_ / BN_;          // 12
    const int tilesM = S_ / BM_;          // 4
    const int perB   = tilesM * tilesN;   // 48
    const int t      = blockIdx.x;        // 192 blocks
    const int b      = t / perB;
    const int rem    = t % perB;
    const int row0   = (rem / tilesN) * BM_;
    const int col0   = (rem % tilesN) * BN_;
    const float* A   = probs + (size_t)b * S_ * S_;
    const float* Bm  = Hi    + (size_t)b * S_ * H_;

    const int tid  = threadIdx.x;
    const int wave = tid >> 5;
    const int lane = tid & 31;
    const int half = lane >> 4, r = lane & 15;
    const int wrow = wave * 16;

    v8f c[4];
    #pragma unroll
    for (int nt = 0; nt < 4; ++nt)
        #pragma unroll
        for (int i = 0; i < 8; ++i) c[nt][i] = 0.f;

    for (int kb = 0; kb < S_; kb += KC_) {
        __syncthreads();
        for (int i = tid; i < BM_ * KC_; i += 256) {
            int rr = i >> 5, kc = i & (KC_ - 1);
            As[rr * LDA_ + kc] = A[(size_t)(row0 + rr) * S_ + kb + kc];
        }
        for (int i = tid; i < KC_ * BN_; i += 256) {
            int kc = i >> 6, n = i & (BN_ - 1);
            Bs[kc * LDB_ + n] = Bm[(size_t)(kb + kc) * H_ + col0 + n];
        }
        __syncthreads();
        #pragma unroll
        for (int kc = 0; kc < KC_; kc += 4) {
            const int kk = kc + 2 * half;
            v2f a;
            a.x = As[(wrow + r) * LDA_ + kk];
            a.y = As[(wrow + r) * LDA_ + kk + 1];
            #pragma unroll
            for (int nt = 0; nt < 4; ++nt) {
                v2f bb;
                bb.x = Bs[kk * LDB_ + nt * 16 + r];
                bb.y = Bs[(kk + 1) * LDB_ + nt * 16 + r];
                c[nt] = wmma_f32(a, bb, c[nt]);
            }
        }
    }
    float* C = ctx + (size_t)b * S_ * H_;
    #pragma unroll
    for (int nt = 0; nt < 4; ++nt)
        #pragma unroll
        for (int i = 0; i < 8; ++i)
            C[(size_t)(row0 + wrow + i + 8 * half) * H_ + col0 + nt * 16 + r] = c[nt][i];
}

// ---------------------------------------------------------------------------
// Kernel 5: h1 = relu([ctx, H_j, ctx*H_j] @ v_w1^T + v_b1)
// rows=2048, K=2304, N=768; virtual X materialized during the LDS stage.
// ---------------------------------------------------------------------------
__device__ __forceinline__ float xval(const float* __restrict__ ctx,
                                      const float* __restrict__ Hj,
                                      int row, int kk) {
    if (kk < H_)          return ctx[(size_t)row * H_ + kk];
    else if (kk < 2 * H_) return Hj[(size_t)row * H_ + kk - H_];
    else { int h = kk - 2 * H_;
           return ctx[(size_t)row * H_ + h] * Hj[(size_t)row * H_ + h]; }
}

__global__ void __launch_bounds__(256)
k_h1(const float* __restrict__ ctx, const float* __restrict__ Hj,
     const float* __restrict__ vw1, const float* __restrict__ vb1,
     float* __restrict__ h1) {
    __shared__ float As[BM_ * LDA_];
    __shared__ float Bs[KC_ * LDB_];

    const int tilesN = OUTH_ / BN_;       // 12
    const int t      = blockIdx.x;        // 16*12 = 192 blocks
    const int row0   = (t / tilesN) * BM_;
    const int col0   = (t % tilesN) * BN_;

    const int tid  = threadIdx.x;
    const int wave = tid >> 5;
    const int lane = tid & 31;
    const int half = lane >> 4, r = lane & 15;
    const int wrow = wave * 16;

    v8f c[4];
    #pragma unroll
    for (int nt = 0; nt < 4; ++nt)
        #pragma unroll
        for (int i = 0; i < 8; ++i) c[nt][i] = 0.f;

    for (int kb = 0; kb < K3H_; kb += KC_) {
        __syncthreads();
        for (int i = tid; i < BM_ * KC_; i += 256) {
            int rr = i >> 5, kc = i & (KC_ - 1);
            As[rr * LDA_ + kc] = xval(ctx, Hj, row0 + rr, kb + kc);
        }
        for (int i = tid; i < KC_ * BN_; i += 256) {
            int n = i >> 5, kc = i & (KC_ - 1);   // coalesced along K of v_w1
            Bs[kc * LDB_ + n] = vw1[(size_t)(col0 + n) * K3H_ + kb + kc];
        }
        __syncthreads();
        #pragma unroll
        for (int kc = 0; kc < KC_; kc += 4) {
            const int kk = kc + 2 * half;
            v2f a;
            a.x = As[(wrow + r) * LDA_ + kk];
            a.y = As[(wrow + r) * LDA_ + kk + 1];
            #pragma unroll
            for (int nt = 0; nt < 4; ++nt) {
                v2f bb;
                bb.x = Bs[kk * LDB_ + nt * 16 + r];
                bb.y = Bs[(kk + 1) * LDB_ + nt * 16 + r];
                c[nt] = wmma_f32(a, bb, c[nt]);
            }
        }
    }
    #pragma unroll
    for (int nt = 0; nt < 4; ++nt)
        #pragma unroll
        for (int i = 0; i < 8; ++i) {
            const int row = row0 + wrow + i + 8 * half;
            const int col = col0 + nt * 16 + r;
            float v = c[nt][i] + vb1[col];
            h1[(size_t)row * OUTH_ + col] = v > 0.f ? v : 0.f;
        }
}

// ---------------------------------------------------------------------------
// Kernel 6: out = alpha * (h1 @ v_w2^T + v_b2)   rows=2048, K=768, N=768
// ---------------------------------------------------------------------------
__global__ void __launch_bounds__(256)
k_out(const float* __restrict__ h1, const float* __restrict__ vw2,
      const float* __restrict__ vb2, const float* __restrict__ alpha,
      float* __restrict__ out) {
    __shared__ float As[BM_ * LDA_];
    __shared__ float Bs[KC_ * LDB_];

    const int tilesN = H_ / BN_;          // 12
    const int t      = blockIdx.x;        // 192 blocks
    const int row0   = (t / tilesN) * BM_;
    const int col0   = (t % tilesN) * BN_;

    const int tid  = threadIdx.x;
    const int wave = tid >> 5;
    const int lane = tid & 31;
    const int half = lane >> 4, r = lane & 15;
    const int wrow = wave * 16;
    const float al = alpha[0];

    v8f c[4];
    #pragma unroll
    for (int nt = 0; nt < 4; ++nt)
        #pragma unroll
        for (int i = 0; i < 8; ++i) c[nt][i] = 0.f;

    for (int kb = 0; kb < OUTH_; kb += KC_) {
        __syncthreads();
        for (int i = tid; i < BM_ * KC_; i += 256) {
            int rr = i >> 5, kc = i & (KC_ - 1);
            As[rr * LDA_ + kc] = h1[(size_t)(row0 + rr) * OUTH_ + kb + kc];
        }
        for (int i = tid; i < KC_ * BN_; i += 256) {
            int n = i >> 5, kc = i & (KC_ - 1);
            Bs[kc * LDB_ + n] = vw2[(size_t)(col0 + n) * OUTH_ + kb + kc];
        }
        __syncthreads();
        #pragma unroll
        for (int kc = 0; kc < KC_; kc += 4) {
            const int kk = kc + 2 * half;
            v2f a;
            a.x = As[(wrow + r) * LDA_ + kk];
            a.y = As[(wrow + r) * LDA_ + kk + 1];
            #pragma unroll
            for (int nt = 0; nt < 4; ++nt) {
                v2f bb;
                bb.x = Bs[kk * LDB_ + nt * 16 + r];
                bb.y = Bs[(kk + 1) * LDB_ + nt * 16 + r];
                c[nt] = wmma_f32(a, bb, c[nt]);
            }
        }
    }
    #pragma unroll
    for (int nt = 0; nt < 4; ++nt)
        #pragma unroll
        for (int i = 0; i < 8; ++i) {
            const int row = row0 + wrow + i + 8 * half;
            const int col = col0 + nt * 16 + r;
            out[(size_t)row * H_ + col] = al * (c[nt][i] + vb2[col]);
        }
}

// ---------------------------------------------------------------------------
extern "C" void kernel_launch(void* const* d_in, const int* in_sizes, int n_in,
                              void* d_out, int out_size, void* d_ws, size_t ws_size,
                              hipStream_t stream) {
    (void)in_sizes; (void)n_in; (void)out_size; (void)ws_size;

    const float* Hj    = (const float*)d_in[0];
    const float* Hi    = (const float*)d_in[1];
    const float* amask = (const float*)d_in[2];
    const float* pjw   = (const float*)d_in[3];
    const float* piw   = (const float*)d_in[4];
    const float* sw1   = (const float*)d_in[5];
    const float* sb1   = (const float*)d_in[6];
    const float* sw2   = (const float*)d_in[7];
    const float* sb2   = (const float*)d_in[8];
    const float* vw1   = (const float*)d_in[9];
    const float* vb1   = (const float*)d_in[10];
    const float* vw2   = (const float*)d_in[11];
    const float* vb2   = (const float*)d_in[12];
    const float* alpha = (const float*)d_in[13];
    float* out = (float*)d_out;

    float* ws    = (float*)d_ws;
    float* Zj    = ws;                           // 49152
    float* Zi    = Zj + (size_t)ROWS_ * D_;      // 49152
    float* aj    = Zi + (size_t)ROWS_ * D_;      // 196608
    float* bi    = aj + (size_t)ROWS_ * M_;      // 196608
    float* probs = bi + (size_t)ROWS_ * M_;      // 1048576
    float* ctx   = probs + (size_t)B_ * S_ * S_; // 1572864
    float* h1    = ctx + (size_t)ROWS_ * H_;     // 1572864

    {
        int total = 2 * ROWS_ * D_;
        k_proj<<<(total + 255) / 256, 256, 0, stream>>>(Hj, Hi, pjw, piw, Zj, Zi);
    }
    {
        int total = 2 * ROWS_ * M_;
        k_ab<<<(total + 255) / 256, 256, 0, stream>>>(Zj, Zi, sw1, aj, bi);
    }
    k_score<<<ROWS_, 256, 0, stream>>>(Zj, Zi, aj, bi, sw1, sb1, sw2, sb2, amask, probs);
    k_ctx<<<B_ * (S_ / BM_) * (H_ / BN_), 256, 0, stream>>>(probs, Hi, ctx);
    k_h1 <<<(ROWS_ / BM_) * (OUTH_ / BN_), 256, 0, stream>>>(ctx, Hj, vw1, vb1, h1);
    k_out<<<(ROWS_ / BM_) * (H_ / BN_), 256, 0, stream>>>(h1, vw2, vb2, alpha, out);
}